// Hysteresis_54288386621868
// MI455X (gfx1250) — compile-verified
//
#include <hip/hip_runtime.h>
#include <hip/hip_bf16.h>
#include <math.h>

// Preisach hysteresis: N=512 grid, T=1024 steps.
// Stage 1: state-resident simulation, per-t block partial sums -> d_ws [T, NBLOCKS].
// Stage 2: cross-block reduction via V_WMMA_F32_16X16X4_F32 (B = ones), then
//          scale * (sum / VEC_LEN) + offset -> d_out [T].

#define NMESH 512
#define TSTEPS 1024
#define VECLEN 131328          // 512*513/2
#define NBLOCKS 128
#define TPB 256
#define CPT 8                  // cells per thread; 128*256*8 == 512*512

typedef float v2f __attribute__((ext_vector_type(2)));
typedef float v8f __attribute__((ext_vector_type(8)));

__device__ __forceinline__ float softplus_f(float x) {
    // numerically stable log(1 + exp(x))
    return fmaxf(x, 0.0f) + log1pf(expf(-fabsf(x)));
}

__global__ __launch_bounds__(TPB) void hyst_stage1(
    const float* __restrict__ h_data,
    const float* __restrict__ raw,
    float* __restrict__ partials)   // [TSTEPS, NBLOCKS]
{
    __shared__ float sh[TSTEPS];
    __shared__ float wsum[TPB / 32];

    const int tid = threadIdx.x;
    for (int i = tid; i < TSTEPS; i += TPB) sh[i] = h_data[i];
    __syncthreads();

    // Each thread owns 8 consecutive cells of one row (512 % 8 == 0).
    const int g    = blockIdx.x * TPB + tid;
    const int base = g * CPT;
    const int row  = base >> 9;          // / 512
    const int col0 = base & (NMESH - 1); // % 512
    const float inv = 1.0f / (float)(NMESH - 1);
    const float alpha = (float)row * inv;
    const int tribase = (row * (row + 1)) >> 1;

    float beta[CPT], dens[CPT], st[CPT];
#pragma unroll
    for (int m = 0; m < CPT; ++m) {
        const int j = col0 + m;
        beta[m] = (float)j * inv;
        dens[m] = (j <= row) ? softplus_f(raw[tribase + j]) : 0.0f;
        st[m]   = -1.0f;                 // initial state
    }

    const int lane = tid & 31;
    const int wave = tid >> 5;

    float h_prev = 0.0f;                 // hs[0] = 0
    for (int t = 0; t < TSTEPS; ++t) {
        const float h_cur = sh[t];
        if (h_cur > h_prev) {
            // up-sweep: alpha < h -> +1 (alpha uniform per thread)
            if (alpha < h_cur) {
#pragma unroll
                for (int m = 0; m < CPT; ++m) st[m] = 1.0f;
            }
        } else if (h_cur < h_prev) {
            // down-sweep: beta > h -> -1
#pragma unroll
            for (int m = 0; m < CPT; ++m)
                if (beta[m] > h_cur) st[m] = -1.0f;
        }

        float acc = 0.0f;
#pragma unroll
        for (int m = 0; m < CPT; ++m) acc = fmaf(st[m], dens[m], acc);

        // wave32 butterfly reduction
#pragma unroll
        for (int off = 16; off >= 1; off >>= 1)
            acc += __shfl_xor(acc, off, 32);

        if (lane == 0) wsum[wave] = acc;
        __syncthreads();
        if (tid == 0) {
            float s = 0.0f;
#pragma unroll
            for (int w = 0; w < TPB / 32; ++w) s += wsum[w];
            partials[t * NBLOCKS + blockIdx.x] = s;
        }
        __syncthreads();
        h_prev = h_cur;
    }
}

// One wave (32 threads) per block, 16 timesteps per block.
// out[t] = scale * ((sum_b partials[t,b]) / VECLEN) + offset
__global__ __launch_bounds__(32) void hyst_stage2(
    const float* __restrict__ partials,
    const float* __restrict__ scale,
    const float* __restrict__ offset,
    float* __restrict__ out)
{
    const int lane = threadIdx.x;    // 0..31, all active -> EXEC all ones
    const int t0   = blockIdx.x * 16;
    const int tl   = lane & 15;      // A-matrix M index (timestep within tile)
    const int kh   = lane >> 4;      // lanes 16-31 carry K=2,3

    v8f c = {};                      // f32 16x16 accumulator, zero-init
    v2f ones;
    ones.x = 1.0f; ones.y = 1.0f;    // all-ones B (4x16) -- layout invariant

    for (int gb = 0; gb < NBLOCKS; gb += 4) {
        const int colk = gb + 2 * kh;
        v2f a;                       // A 16x4 f32 fragment (ISA 7.12.2 layout)
        a.x = partials[(t0 + tl) * NBLOCKS + colk];
        a.y = partials[(t0 + tl) * NBLOCKS + colk + 1];
        c = __builtin_amdgcn_wmma_f32_16x16x4_f32(
                /*neg_a=*/false, a, /*neg_b=*/false, ones,
                /*c_mod=*/(short)0, c, /*reuse_a=*/false, /*reuse_b=*/false);
    }

    const float sc = scale[0];
    const float of = offset[0];
    const float invlen = 1.0f / (float)VECLEN;

    // D column N=0: lane 0 holds M=0..7 in c[0..7], lane 16 holds M=8..15.
    if (lane == 0) {
#pragma unroll
        for (int r = 0; r < 8; ++r)
            out[t0 + r] = fmaf(sc, c[r] * invlen, of);
    } else if (lane == 16) {
#pragma unroll
        for (int r = 0; r < 8; ++r)
            out[t0 + 8 + r] = fmaf(sc, c[r] * invlen, of);
    }
}

extern "C" void kernel_launch(void* const* d_in, const int* in_sizes, int n_in,
                              void* d_out, int out_size, void* d_ws, size_t ws_size,
                              hipStream_t stream) {
    const float* h_data = (const float*)d_in[0];  // [1024]
    const float* raw    = (const float*)d_in[1];  // [131328]
    const float* scale  = (const float*)d_in[2];  // [1]
    const float* offset = (const float*)d_in[3];  // [1]
    float* out = (float*)d_out;                   // [1024]
    float* partials = (float*)d_ws;               // [1024 * 128] floats = 512 KB

    hyst_stage1<<<NBLOCKS, TPB, 0, stream>>>(h_data, raw, partials);
    hyst_stage2<<<TSTEPS / 16, 32, 0, stream>>>(partials, scale, offset, out);
}